// PostprocessingLayer_17927193494104
// MI455X (gfx1250) — compile-verified
//
#include <hip/hip_runtime.h>
#include <cstdint>
#include <cstddef>

// Problem geometry (reference: y (64, 80, 80, 84), IW=IH=320, K=100)
#define BATCH 64
#define HW    80          // H == W
#define CH    80          // heatmap channels C
#define CP    84          // C + 4 (wh, bc)
#define KTOP  100
#define ROWKEYS   25      // 80*80 / 256
#define MERGEKEYS 32      // ceil(80*100 / 256)
#define NCAND_PER_B (HW * KTOP)   // 8000 candidates per batch

typedef unsigned int v4u __attribute__((ext_vector_type(4)));
typedef int          v8i __attribute__((ext_vector_type(8)));
typedef int          v4i __attribute__((ext_vector_type(4)));

// ---------------------------------------------------------------------------
// 64-bit sort key: monotonic(score) in high 32 bits, ~idx in low 32 bits.
// Larger key == larger score; equal scores -> smaller idx wins (stable argsort).
// ---------------------------------------------------------------------------
__device__ __forceinline__ unsigned long long pack_key(float s, unsigned idx) {
  unsigned u = __float_as_uint(s);
  unsigned mono = (u & 0x80000000u) ? ~u : (u | 0x80000000u);
  return ((unsigned long long)mono << 32) | (unsigned)(~idx);
}

// ---------------------------------------------------------------------------
// Exact block-wide "want-th largest" of unique 64-bit keys via 8 passes of
// 8-bit LDS-histogram radix select. 256 threads, 256 bins (bin == tid).
// The degenerate suppressed-zero class (mono == 0x80000000, thousands of keys
// sharing the same digit in the 4 mono passes) is aggregated per-thread to a
// single atomicAdd to avoid a same-address LDS atomic pileup.
// ---------------------------------------------------------------------------
template <int NK>
__device__ unsigned long long block_radix_kth(const unsigned long long (&lk)[NK],
                                              int want,
                                              int* hist, int* seg,
                                              int* selBin, int* selAbove) {
  const int tid  = threadIdx.x;
  const int lane = tid & 31;
  const int wid  = tid >> 5;
  unsigned long long prefixKey = 0ull, remMask = 0ull;
  const unsigned long long zkeyHi = 0x80000000ull << 32;   // +0.0 score class

  for (int pass = 0; pass < 8; ++pass) {
    const int shift = 56 - 8 * pass;
    const bool monoPass = (pass < 4);
    hist[tid] = 0;
    __syncthreads();

    int zcnt = 0;
#pragma unroll
    for (int e = 0; e < NK; ++e) {
      const unsigned long long kk = lk[e];
      if ((kk & remMask) != prefixKey) continue;
      if (monoPass && ((unsigned)(kk >> 32) == 0x80000000u)) { ++zcnt; continue; }
      atomicAdd(&hist[(unsigned)(kk >> shift) & 255u], 1);
    }
    if (monoPass && zcnt && ((zkeyHi & remMask) == prefixKey)) {
      const unsigned zdig = (pass == 0) ? 0x80u : 0x00u;   // digits of +0.0 key
      atomicAdd(&hist[zdig], zcnt);
    }
    __syncthreads();

    // Two-level inclusive suffix scan: cge[t] = sum(hist[t..255])
    int v = hist[tid];
#pragma unroll
    for (int off = 1; off < 32; off <<= 1) {
      int o = __shfl_down(v, off, 32);
      if (lane + off < 32) v += o;
    }
    if (lane == 0) seg[wid] = v;            // per-wave segment totals
    __syncthreads();
    if (tid < 8) {                          // exclusive suffix over 8 segments
      int acc = 0;
      for (int q = tid + 1; q < 8; ++q) acc += seg[q];
      seg[8 + tid] = acc;
    }
    __syncthreads();
    const int cge = v + seg[8 + wid];
    hist[tid] = cge;
    __syncthreads();
    const int cgeNext = (tid < 255) ? hist[tid + 1] : 0;
    if (cge >= want && cgeNext < want) { *selBin = tid; *selAbove = cgeNext; }
    __syncthreads();

    prefixKey |= ((unsigned long long)(unsigned)(*selBin)) << shift;
    remMask   |= (0xFFull << shift);
    want      -= *selAbove;
    __syncthreads();
  }
  return prefixKey;   // exact want-th largest key (keys are unique)
}

// ---------------------------------------------------------------------------
// Phase 1: one block per (batch b, row i). TDM-load 3 heatmap rows (channels
// 0..79 of the 84-ch rows, strided 3-D tile) into LDS, 3x3 NMS, per-row exact
// top-100 -> candidate keys in workspace.
// ---------------------------------------------------------------------------
__global__ __launch_bounds__(256) void rownms_topk(const float* __restrict__ y,
                                                   unsigned long long* __restrict__ cand) {
  const int i   = blockIdx.x;    // row 0..79
  const int b   = blockIdx.y;    // batch
  const int tid = threadIdx.x;

  extern __shared__ float smem[];            // 3 * 80 * 80 floats (76.8 KB)
#define TILE(s, j, c) smem[(((s) * HW) + (j)) * CH + (c)]
  __shared__ int hist[256];
  __shared__ int seg[16];
  __shared__ int selBin, selAbove, outCnt;

  const int r0    = (i == 0) ? 0 : (i - 1);
  const int r1    = (i == HW - 1) ? (HW - 1) : (i + 1);
  const int nrows = r1 - r0 + 1;
  const int s0    = (i == 0) ? 1 : 0;        // LDS slot of row r0

#if __has_builtin(__builtin_amdgcn_tensor_load_to_lds)
  if (tid < 32) {   // wave 0 issues one TDM descriptor for the whole 3-row tile
    const size_t gElem = (((size_t)b * HW + (size_t)r0) * HW) * CP;
    const unsigned long long ga = (unsigned long long)(uintptr_t)(y + gElem);
    const unsigned ldsA = (unsigned)(size_t)&TILE(s0, 0, 0);

    v4u g0;
    g0.x = 1u;                                  // count=1, user mode, no gather
    g0.y = ldsA;                                // lds_addr (bytes)
    g0.z = (unsigned)ga;                        // global_addr[31:0]
    g0.w = (unsigned)(ga >> 32) | (2u << 30);   // global_addr[56:32] | type=2

    v8i g1;
    g1[0] = (int)(2u << 16);                    // data_size = 4 bytes
    g1[1] = (int)((unsigned)CP << 16);          // tensor_dim0 = 84
    g1[2] = (int)((unsigned)HW << 16);          // tensor_dim1 = 80
    g1[3] = (int)((unsigned)CH << 16);          // tile_dim0   = 80 (channels)
    g1[4] = (int)((unsigned)HW | ((unsigned)nrows << 16)); // tile_dim1=80, tile_dim2=nrows
    g1[5] = (int)CP;                            // tensor_dim0_stride = 84
    g1[6] = (int)(((unsigned)(HW * CP) & 0xFFFFu) << 16);  // tensor_dim1_stride = 6720
    g1[7] = 0;

    v4i g2;
    g2[0] = HW;                                 // tensor_dim2 = 80
    g2[1] = 0;                                  // tensor_dim3 unused
    g2[2] = HW * HW * CP;                       // tensor_dim2_stride = 537600
    g2[3] = 0;                                  // tile_dim3 = 0
    v4i g3 = {0, 0, 0, 0};
    v8i g4 = {0, 0, 0, 0, 0, 0, 0, 0};          // extra operand (clang-23 form), zero

    __builtin_amdgcn_tensor_load_to_lds(g0, g1, g2, g3, g4, 0);
    __builtin_amdgcn_s_wait_tensorcnt(0);
  }
#else
  for (int p = tid; p < nrows * HW * CH; p += 256) {
    const int c = p % CH, rest = p / CH;
    const int j = rest % HW, rr = rest / HW;
    TILE(s0 + rr, j, c) = y[(((size_t)b * HW + (r0 + rr)) * HW + j) * CP + c];
  }
#endif

  // -inf pad for the missing halo row at image borders (SAME padding)
  if (i == 0)
    for (int p = tid; p < HW * CH; p += 256) smem[p] = -__builtin_huge_valf();
  if (i == HW - 1)
    for (int p = tid; p < HW * CH; p += 256) smem[2 * HW * CH + p] = -__builtin_huge_valf();
  if (tid == 0) outCnt = 0;
  __syncthreads();

  // 3x3 NMS -> packed keys (25 per thread, coalesced over channels)
  unsigned long long lk[ROWKEYS];
#pragma unroll
  for (int e = 0; e < ROWKEYS; ++e) {
    const int p = tid + e * 256;               // < 6400
    const int c = p % CH;
    const int j = p / CH;
    const float v = TILE(1, j, c);
    float m = v;
#pragma unroll
    for (int s = 0; s < 3; ++s) {
      m = fmaxf(m, TILE(s, j, c));
      if (j > 0)      m = fmaxf(m, TILE(s, j - 1, c));
      if (j < HW - 1) m = fmaxf(m, TILE(s, j + 1, c));
    }
    const float keep  = (v == m) ? v : 0.0f;   // keep = hmax * (hm == hmax)
    const unsigned idx = (unsigned)(((i * HW) + j) * CH + c);
    lk[e] = pack_key(keep, idx);
  }

  const unsigned long long thr =
      block_radix_kth<ROWKEYS>(lk, KTOP, hist, seg, &selBin, &selAbove);

  // Compact exactly the top-100 keys of this row (unordered)
  unsigned long long* dst = cand + ((size_t)b * HW + (size_t)i) * KTOP;
#pragma unroll
  for (int e = 0; e < ROWKEYS; ++e) {
    if (lk[e] >= thr) {
      const int pos = atomicAdd(&outCnt, 1);
      dst[pos] = lk[e];
    }
  }
#undef TILE
}

// ---------------------------------------------------------------------------
// Phase 2: one block per batch. Merge 8000 candidates -> exact top-100,
// rank-order, decode, gather bc / exp(wh)-1, write all four outputs.
// ---------------------------------------------------------------------------
__global__ __launch_bounds__(256) void merge_topk(const float* __restrict__ y,
                                                  const unsigned long long* __restrict__ cand,
                                                  float* __restrict__ out) {
  const int b   = blockIdx.x;
  const int tid = threadIdx.x;

  __shared__ int hist[256];
  __shared__ int seg[16];
  __shared__ int selBin, selAbove, outCnt;
  __shared__ unsigned long long win[KTOP];

  unsigned long long lk[MERGEKEYS];
  const unsigned long long* base = cand + (size_t)b * NCAND_PER_B;
#pragma unroll
  for (int e = 0; e < MERGEKEYS; ++e) {
    const int g = tid + e * 256;
    lk[e] = (g < NCAND_PER_B) ? base[g] : 0ull;   // sentinel 0 < any real key
  }
  if (tid == 0) outCnt = 0;

  const unsigned long long thr =
      block_radix_kth<MERGEKEYS>(lk, KTOP, hist, seg, &selBin, &selAbove);

#pragma unroll
  for (int e = 0; e < MERGEKEYS; ++e) {
    if (lk[e] >= thr) {
      const int pos = atomicAdd(&outCnt, 1);
      win[pos] = lk[e];
    }
  }
  __syncthreads();

  if (tid < KTOP) {
    const unsigned long long mk = win[tid];
    int rank = 0;
    for (int q = 0; q < KTOP; ++q) rank += (win[q] > mk) ? 1 : 0;  // keys unique

    const unsigned idx   = ~(unsigned)mk;
    const unsigned mono  = (unsigned)(mk >> 32);
    const unsigned ubits = (mono & 0x80000000u) ? (mono & 0x7FFFFFFFu) : ~mono;
    const float score = __uint_as_float(ubits);

    const int c  = (int)(idx % CH);
    const int j  = (int)((idx / CH) % HW);
    const int ii = (int)(idx / (CH * HW));

    const float* row = y + (((size_t)b * HW + ii) * HW + j) * CP;
    const float bcx = row[CH + 2];
    const float bcy = row[CH + 3];
    const float whx = expf(row[CH + 0]) - 1.0f;
    const float why = expf(row[CH + 1]) - 1.0f;

    const int o = b * KTOP + rank;
    out[o]                 = score;                 // score_k
    out[BATCH * KTOP + o]  = (float)c;              // class id k
    out[2 * BATCH * KTOP + 2 * o + 0] = 4.0f * (float)j + bcx;   // centers.x
    out[2 * BATCH * KTOP + 2 * o + 1] = 4.0f * (float)ii + bcy;  // centers.y
    out[4 * BATCH * KTOP + 2 * o + 0] = 4.0f * whx;              // wh.x
    out[4 * BATCH * KTOP + 2 * o + 1] = 4.0f * why;              // wh.y
  }
}

// ---------------------------------------------------------------------------
extern "C" void kernel_launch(void* const* d_in, const int* in_sizes, int n_in,
                              void* d_out, int out_size, void* d_ws, size_t ws_size,
                              hipStream_t stream) {
  (void)in_sizes; (void)n_in; (void)out_size; (void)ws_size;
  const float* y = (const float*)d_in[0];
  unsigned long long* cand = (unsigned long long*)d_ws;   // 64*80*100*8 = 4.1 MB
  float* out = (float*)d_out;

  dim3 grid1(HW, BATCH);
  const size_t tileBytes = (size_t)3 * HW * CH * sizeof(float);  // 76.8 KB LDS
  hipLaunchKernelGGL(rownms_topk, grid1, dim3(256), tileBytes, stream, y, cand);
  hipLaunchKernelGGL(merge_topk, dim3(BATCH), dim3(256), 0, stream, y, cand, out);
}